// anticipation_mae_82162724372813
// MI455X (gfx1250) — compile-verified
//
#include <hip/hip_runtime.h>
#include <stdint.h>

// ---------------------------------------------------------------------------
// anticipation-MAE reduction for MI455X (gfx1250).
// Memory-bound: 128 MiB streamed once -> ~5.8us floor at 23.3 TB/s.
// Data path: per-wave double-buffered async global->LDS (ASYNCcnt) with
// non-temporal hints (read-once stream), wave32 butterfly reductions,
// deterministic two-pass aggregation (no atomics).
// ---------------------------------------------------------------------------

#define HVAL  7500.0f
#define H09   6750.0f   // 0.9*H exact in f32
#define H01    750.0f   // 0.1*H exact in f32

#define N_DIM   128
#define T_DIM   16384
#define C_DIM   8
#define CHUNKS  8                          // T-chunks per n
#define NBLOCKS (N_DIM * CHUNKS)           // 1024 blocks
#define BLOCK_THREADS 256                  // 8 waves of 32
#define WAVES   8
#define ITERS   16                         // per-wave iterations
// per block: 2048 rows * 8 ch * 4B = 64 KiB per input
#define BLOCK_BYTES 65536u

// ---- CDNA5 async global->LDS helpers (inline asm; portable per bridge doc) --
__device__ __forceinline__ uint32_t lds_off_of(const void* p) {
  // generic shared pointer: addr[31:0] is the LDS byte offset
  return (uint32_t)(uintptr_t)p;
}

__device__ __forceinline__ void async_b128_nt(uint32_t lds_off, uint32_t goff,
                                              const void* sbase) {
  // read-once stream: non-temporal hint so 128 MiB doesn't rinse the caches
  asm volatile("global_load_async_to_lds_b128 %0, %1, %2 th:TH_LOAD_NT"
               :
               : "v"(lds_off), "v"(goff), "s"(sbase)
               : "memory");
}
__device__ __forceinline__ void wait_async_le2() {
  asm volatile("s_wait_asynccnt 2" ::: "memory");
}
__device__ __forceinline__ void wait_async_0() {
  asm volatile("s_wait_asynccnt 0" ::: "memory");
}
__device__ __forceinline__ void wait_ds_0() {
  // WAR guard: DS reads of the buffer being re-targeted must be complete
  // before the async engine may overwrite it (DS and ASYNC queues are
  // mutually unordered). These waits are ~free: the DS ops they cover
  // finished a full iteration earlier.
  asm volatile("s_wait_dscnt 0" ::: "memory");
}

// ---- per-element masked accumulation (8 quantities) -------------------------
__device__ __forceinline__ void accum1(float* a, float ov, float tv) {
  float oc = (ov > HVAL || ov < 0.0f) ? HVAL : ov;
  float tc = (tv > HVAL || tv < 0.0f) ? HVAL : tv;
  float d  = fabsf(oc - tc);
  bool mo = (tc == HVAL);
  bool mi = (tc < HVAL) && (tc > 0.0f);
  bool mp = (oc < H09) && (oc > H01);
  bool me = (tc < H01) && (tc > 0.0f);
  a[0] += mo ? d : 0.0f;  a[1] += mo ? 1.0f : 0.0f;
  a[2] += mi ? d : 0.0f;  a[3] += mi ? 1.0f : 0.0f;
  a[4] += mp ? d : 0.0f;  a[5] += mp ? 1.0f : 0.0f;
  a[6] += me ? d : 0.0f;  a[7] += me ? 1.0f : 0.0f;
}

// ---------------------------------------------------------------------------
// Kernel 1: per-(block, channel) partial sums/counts of the 4 masked streams.
// ---------------------------------------------------------------------------
__global__ __launch_bounds__(BLOCK_THREADS)
void k1_partials(const float* __restrict__ out_rsd,
                 const float* __restrict__ tgt_rsd,
                 float* __restrict__ ws) {
  // stage[wave][phase][input][128 floats]  (512 B per tile) = 16 KiB
  __shared__ float stage[WAVES][2][2][128];
  __shared__ float partial[WAVES * 64];

  const int tid  = threadIdx.x;
  const int w    = tid >> 5;
  const int lane = tid & 31;
  const int par  = lane & 1;      // float4-index parity -> channel group 0..3 / 4..7

  const char* ob = (const char*)out_rsd + (size_t)blockIdx.x * BLOCK_BYTES;
  const char* tb = (const char*)tgt_rsd + (size_t)blockIdx.x * BLOCK_BYTES;

  // per-lane byte offset within the block's 64 KiB region
  const uint32_t vbase = (uint32_t)(w * 8192 + lane * 16);
  const uint32_t lbyte = (uint32_t)(lane * 16);

  uint32_t ls_o[2], ls_t[2];
  ls_o[0] = lds_off_of(&stage[w][0][0][0]) + lbyte;
  ls_o[1] = lds_off_of(&stage[w][1][0][0]) + lbyte;
  ls_t[0] = lds_off_of(&stage[w][0][1][0]) + lbyte;
  ls_t[1] = lds_off_of(&stage[w][1][1][0]) + lbyte;

  float acc[4][8];
#pragma unroll
  for (int j = 0; j < 4; ++j)
#pragma unroll
    for (int q = 0; q < 8; ++q) acc[j][q] = 0.0f;

  // prologue: issue batch 0
  async_b128_nt(ls_o[0], vbase, ob);
  async_b128_nt(ls_t[0], vbase, tb);

  for (int it = 0; it < ITERS; ++it) {
    const int nb = it + 1;
    if (nb < ITERS) {
      const uint32_t goff = vbase + (uint32_t)nb * 512u;
      wait_ds_0();                // buffer (nb&1) fully consumed by DS reads
      async_b128_nt(ls_o[nb & 1], goff, ob);
      async_b128_nt(ls_t[nb & 1], goff, tb);
      wait_async_le2();           // batch `it` landed in LDS
    } else {
      wait_async_0();
    }
    const int ph = it & 1;
    float4 o4 = *(const float4*)&stage[w][ph][0][lane * 4];
    float4 t4 = *(const float4*)&stage[w][ph][1][lane * 4];
    accum1(acc[0], o4.x, t4.x);
    accum1(acc[1], o4.y, t4.y);
    accum1(acc[2], o4.z, t4.z);
    accum1(acc[3], o4.w, t4.w);
  }

  // butterfly over same-parity lanes (xor masks keep bit0 -> channels stay split)
#pragma unroll
  for (int s = 16; s >= 2; s >>= 1)
#pragma unroll
    for (int j = 0; j < 4; ++j)
#pragma unroll
      for (int q = 0; q < 8; ++q)
        acc[j][q] += __shfl_xor(acc[j][q], s, 32);

  if (lane < 2) {                 // lane == parity here
#pragma unroll
    for (int j = 0; j < 4; ++j)
#pragma unroll
      for (int q = 0; q < 8; ++q)
        partial[w * 64 + (4 * par + j) * 8 + q] = acc[j][q];
  }
  __syncthreads();

  if (tid < 64) {                 // tid = channel*8 + quantity
    float s = 0.0f;
#pragma unroll
    for (int ww = 0; ww < WAVES; ++ww) s += partial[ww * 64 + tid];
    ws[(size_t)blockIdx.x * 64 + tid] = s;
  }
}

// ---------------------------------------------------------------------------
// Kernel 2: combine chunk partials -> per-cell means -> 4 global aggregates.
// ---------------------------------------------------------------------------
__global__ __launch_bounds__(1024)
void k2_finalize(const float* __restrict__ ws, float* __restrict__ out) {
  __shared__ float red[32 * 8];
  __shared__ float tot[8];

  const int tid = threadIdx.x;      // 0..1023 -> cell (n, c)
  const int n = tid >> 3;
  const int c = tid & 7;

  float q[8];
#pragma unroll
  for (int k = 0; k < 8; ++k) q[k] = 0.0f;
  for (int ch = 0; ch < CHUNKS; ++ch) {
    const float* pp = ws + ((size_t)(n * CHUNKS + ch)) * 64 + c * 8;
#pragma unroll
    for (int k = 0; k < 8; ++k) q[k] += pp[k];
  }

  const float oS = q[0], oC = q[1], iS = q[2], iC = q[3];
  const float pS = q[4], pC = q[5], eS = q[6], eC = q[7];
  const float oM = oS / fmaxf(oC, 1.0f);
  const float iM = iS / fmaxf(iC, 1.0f);
  const float pM = pS / fmaxf(pC, 1.0f);
  const float eM = eS / fmaxf(eC, 1.0f);
  const bool ov = oC > 0.0f, iv = iC > 0.0f, pv = pC > 0.0f, ev = eC > 0.0f;
  const bool wvld = ov || iv;
  const float wM = (ov && iv) ? 0.5f * (oM + iM) : (ov ? oM : iM);

  float r[8];
  r[0] = wvld ? wM : 0.0f;  r[1] = wvld ? 1.0f : 0.0f;
  r[2] = iv ? iM : 0.0f;    r[3] = iv ? 1.0f : 0.0f;
  r[4] = pv ? pM : 0.0f;    r[5] = pv ? 1.0f : 0.0f;
  r[6] = ev ? eM : 0.0f;    r[7] = ev ? 1.0f : 0.0f;

#pragma unroll
  for (int s = 16; s >= 1; s >>= 1)
#pragma unroll
    for (int k = 0; k < 8; ++k) r[k] += __shfl_xor(r[k], s, 32);

  const int wid = tid >> 5, lane = tid & 31;
  if (lane == 0) {
#pragma unroll
    for (int k = 0; k < 8; ++k) red[wid * 8 + k] = r[k];
  }
  __syncthreads();

  if (tid < 8) {
    float s = 0.0f;
#pragma unroll
    for (int ww = 0; ww < 32; ++ww) s += red[ww * 8 + tid];
    tot[tid] = s;
  }
  __syncthreads();

  if (tid == 0) {
    out[0] = tot[0] / tot[1];   // wMAE
    out[1] = tot[2] / tot[3];   // inMAE
    out[2] = tot[4] / tot[5];   // pMAE
    out[3] = tot[6] / tot[7];   // eMAE
  }
}

extern "C" void kernel_launch(void* const* d_in, const int* in_sizes, int n_in,
                              void* d_out, int out_size, void* d_ws, size_t ws_size,
                              hipStream_t stream) {
  (void)in_sizes; (void)n_in; (void)out_size; (void)ws_size;
  const float* o = (const float*)d_in[0];
  const float* t = (const float*)d_in[1];
  float* ws  = (float*)d_ws;     // needs 1024*64*4 = 256 KiB
  float* out = (float*)d_out;    // 4 floats

  k1_partials<<<NBLOCKS, BLOCK_THREADS, 0, stream>>>(o, t, ws);
  k2_finalize<<<1, 1024, 0, stream>>>(ws, out);
}